// MultiHeadAdaptiveChebyshevLayer_88278757802344
// MI455X (gfx1250) — compile-verified
//
#include <hip/hip_runtime.h>

typedef float v2f __attribute__((ext_vector_type(2)));
typedef float v8f __attribute__((ext_vector_type(8)));

#define BATCH 1024
#define ISZ   1024
#define NH    8
#define MT    8
#define KS    8
#define FEAT  (NH * ISZ * KS)   /* 65536 */
#define IC    64                /* i-columns per workgroup chunk */
#define BT    16                /* batch rows per tile (WMMA M) */
#define LN_EPS 1e-5f

__global__ void stats_init_kernel(float* __restrict__ stats) {
    int i = blockIdx.x * blockDim.x + threadIdx.x;
    if (i < 2 * BATCH) stats[i] = 0.f;
}

// xor-reduce over each 16-lane half independently (xor masks 1,2,4,8 keep bit4).
// ds_swizzle group-of-32 encoding: offset = {0, xor[4:0], or[4:0]=0, and[4:0]=0x1f}
__device__ __forceinline__ float half_reduce_add(float v) {
#if __has_builtin(__builtin_amdgcn_ds_swizzle)
    v += __int_as_float(__builtin_amdgcn_ds_swizzle(__float_as_int(v), 0x041f));
    v += __int_as_float(__builtin_amdgcn_ds_swizzle(__float_as_int(v), 0x081f));
    v += __int_as_float(__builtin_amdgcn_ds_swizzle(__float_as_int(v), 0x101f));
    v += __int_as_float(__builtin_amdgcn_ds_swizzle(__float_as_int(v), 0x201f));
#else
    v += __shfl_xor(v, 1, 32);
    v += __shfl_xor(v, 2, 32);
    v += __shfl_xor(v, 4, 32);
    v += __shfl_xor(v, 8, 32);
#endif
    return v;
}

// PHASE 0: compute y on the fly, accumulate per-row sum/sumsq into stats[row]/stats[1024+row]
// PHASE 1: recompute y, normalize with stats, single 256MB output write
template <int PHASE>
__global__ __launch_bounds__(256) void cheb_wmma_kernel(
    const float* __restrict__ x,      // [B, I]
    const float* __restrict__ scale,  // [H, I]
    const float* __restrict__ poly,   // [H, I, K]
    const float* __restrict__ kern,   // [H, I, M, K]
    const float* __restrict__ gamma,  // [F]
    const float* __restrict__ beta,   // [F]
    float* __restrict__ out,          // [B, F]
    float* __restrict__ stats)        // [2*B] (d_ws)
{
    __shared__ float xlds[BT * IC];   // 4 KB x tile
    __shared__ float sstat[32];       // 16 rows x {sum, sumsq}

    const int c    = blockIdx.x;      // i-chunk index
    const int bt   = blockIdx.y;      // batch tile index
    const int tid  = threadIdx.x;
    const int lane = tid & 31;
    const int w    = tid >> 5;        // wave id == head id
    const int half = lane >> 4;       // 0: lanes 0-15, 1: lanes 16-31
    const int lr   = lane & 15;       // batch row (A) / k column (B, C/D)
    const int kcol = lr & 7;          // clamped, always-in-bounds k column
    const bool kv  = (lr < KS);       // lanes with a real k column
    const bool hb  = (half != 0);
    const int i0   = c * IC;

    // Stage x tile [BT x IC] into LDS, coalesced.
    for (int e = tid; e < BT * IC; e += 256) {
        int row = e >> 6;             // /IC
        int col = e & (IC - 1);
        xlds[e] = x[(size_t)(bt * BT + row) * ISZ + i0 + col];
    }
    if (PHASE == 0 && tid < 32) sstat[tid] = 0.f;
    __syncthreads();

    // Per-row LN params for the 8 D-rows this lane-half owns (phase 1 only).
    float mean[8], rstd[8];
    if (PHASE == 1) {
#pragma unroll
        for (int r = 0; r < 8; ++r) {
            int row   = bt * BT + r + 8 * half;
            float mu  = stats[row] * (1.f / (float)FEAT);
            float msq = stats[BATCH + row] * (1.f / (float)FEAT);
            mean[r]   = mu;
            rstd[r]   = rsqrtf(msq - mu * mu + LN_EPS);
        }
    }

    float s1[8], s2[8];
#pragma unroll
    for (int r = 0; r < 8; ++r) { s1[r] = 0.f; s2[r] = 0.f; }

    const int h = w;                  // one head per wave

    // Per-lane base pointers (all accesses below are in-bounds for every lane).
    const float* sp = scale + (size_t)h * ISZ + i0;                       // scalar per i
    const float* kp = kern + (size_t)(h * ISZ + i0) * (MT * KS)
                           + half * 16 + kcol;                            // rows {2h,2h+1,2h+4,2h+5}
    const float* pp = poly + (size_t)(h * ISZ + i0) * KS + kcol;
    const float* gp = gamma + (size_t)(h * ISZ + i0) * KS + kcol;
    const float* bp = beta  + (size_t)(h * ISZ + i0) * KS + kcol;
    const size_t outbase = (size_t)(h * ISZ + i0) * KS + kcol;

    for (int t = 0; t < IC; ++t) {
        const float xv = xlds[lr * IC + t] * sp[t];

        // Chebyshev recurrence as individual scalars (no dynamic indexing).
        const float u  = xv + xv;
        const float T0 = 1.f;
        const float T1 = xv;
        const float T2 = __builtin_fmaf(u, T1, -T0);
        const float T3 = __builtin_fmaf(u, T2, -T1);
        const float T4 = __builtin_fmaf(u, T3, -T2);
        const float T5 = __builtin_fmaf(u, T4, -T3);
        const float T6 = __builtin_fmaf(u, T5, -T4);
        const float T7 = __builtin_fmaf(u, T6, -T5);

        // A 16x4: lane = batch row lr; VGPR0/1 carry K-slots {0,2}/{1,3} by half.
        v2f a1, a2;
        a1[0] = hb ? T2 : T0;
        a1[1] = hb ? T3 : T1;
        a2[0] = hb ? T6 : T4;
        a2[1] = hb ? T7 : T5;

        // B 4x16: unconditional clamped loads, then value-select (pure cndmask).
        const float kb0 = kp[0];
        const float kb1 = kp[8];
        const float kb2 = kp[32];
        const float kb3 = kp[40];
        kp += MT * KS;
        v2f b1, b2;
        b1[0] = kv ? kb0 : 0.f;
        b1[1] = kv ? kb1 : 0.f;
        b2[0] = kv ? kb2 : 0.f;
        b2[1] = kv ? kb3 : 0.f;

        // D[16x16] = A*B + C, chained K=4 reductions over m=0..3 then m=4..7.
        v8f d = {0.f, 0.f, 0.f, 0.f, 0.f, 0.f, 0.f, 0.f};
        d = __builtin_amdgcn_wmma_f32_16x16x4_f32(false, a1, false, b1,
                                                  (short)0, d, false, false);
        d = __builtin_amdgcn_wmma_f32_16x16x4_f32(false, a2, false, b2,
                                                  (short)0, d, false, false);

        const float pwl = pp[t * KS];
        const float pw  = kv ? pwl : 0.f;

        if (PHASE == 0) {
#pragma unroll
            for (int r = 0; r < 8; ++r) {
                float y = d[r] * pw;   // lane: (row r+8*half, k=lr); zero for k>=8
                s1[r] += y;
                s2[r] += y * y;
            }
        } else {
            const float g  = gp[t * KS];
            const float be = bp[t * KS];
            if (kv) {                  // single exec toggle for the 8 stores
                const size_t col = outbase + (size_t)t * KS;
#pragma unroll
                for (int r = 0; r < 8; ++r) {
                    float y = d[r] * pw;
                    float o = (y - mean[r]) * rstd[r] * g + be;
                    out[(size_t)(bt * BT + r + 8 * half) * FEAT + col] = o;
                }
            }
        }
    }

    if (PHASE == 0) {
#pragma unroll
        for (int r = 0; r < 8; ++r) {
            s1[r] = half_reduce_add(s1[r]);
            s2[r] = half_reduce_add(s2[r]);
        }
        if (lr == 0) {  // one lane per half flushes its 8 rows to LDS
#pragma unroll
            for (int r = 0; r < 8; ++r) {
                atomicAdd(&sstat[r + 8 * half], s1[r]);
                atomicAdd(&sstat[16 + r + 8 * half], s2[r]);
            }
        }
        __syncthreads();
        if (tid < BT) {  // one global atomic pair per batch row per WG
            atomicAdd(&stats[bt * BT + tid], sstat[tid]);
            atomicAdd(&stats[BATCH + bt * BT + tid], sstat[16 + tid]);
        }
    }
}

extern "C" void kernel_launch(void* const* d_in, const int* in_sizes, int n_in,
                              void* d_out, int out_size, void* d_ws, size_t ws_size,
                              hipStream_t stream) {
    const float* x     = (const float*)d_in[0];
    const float* scale = (const float*)d_in[1];
    const float* poly  = (const float*)d_in[2];
    const float* kern  = (const float*)d_in[3];
    const float* gamma = (const float*)d_in[4];
    const float* beta  = (const float*)d_in[5];
    float* out   = (float*)d_out;
    float* stats = (float*)d_ws;   // 2*BATCH floats of scratch

    stats_init_kernel<<<dim3((2 * BATCH + 255) / 256), dim3(256), 0, stream>>>(stats);

    dim3 grid(ISZ / IC, BATCH / BT);  // 16 x 64 workgroups, 256 threads each
    cheb_wmma_kernel<0><<<grid, dim3(256), 0, stream>>>(x, scale, poly, kern,
                                                        gamma, beta, out, stats);
    cheb_wmma_kernel<1><<<grid, dim3(256), 0, stream>>>(x, scale, poly, kern,
                                                        gamma, beta, out, stats);
}